// ClosestEmbeddingsLayer_60936995996269
// MI455X (gfx1250) — compile-verified
//
#include <hip/hip_runtime.h>
#include <hip/hip_bf16.h>

typedef __attribute__((ext_vector_type(2))) float v2f;
typedef __attribute__((ext_vector_type(8))) float v8f;

#define WAVES_PER_BLOCK 8
#define TOPK 500
#define SORT_N 512

// ---------------------------------------------------------------------------
// Kernel 1: scores[b, v] = dot(gen[b, :], table[v, :])  via V_WMMA_F32_16X16X4_F32
// Each block: 16 output rows (shared A tile in LDS), 8 waves x 16 columns.
// ---------------------------------------------------------------------------
__global__ void gemm_wmma_f32(const float* __restrict__ gen,
                              const float* __restrict__ table,
                              float* __restrict__ scores,
                              int V, int D, int colTiles) {
    __shared__ float Atile[16 * 128];   // D == 128
    const int tid  = threadIdx.x;
    const int wave = tid >> 5;
    const int lane = tid & 31;
    const int half = lane >> 4;         // 0: lanes 0-15, 1: lanes 16-31
    const int l16  = lane & 15;
    const int row0 = blockIdx.y * 16;

    // cooperative load of the 16 x D A tile (shared by all waves in block)
    const float4* gsrc = (const float4*)(gen + (size_t)row0 * D);
    for (int i = tid; i < (16 * D) / 4; i += blockDim.x) {
        ((float4*)Atile)[i] = gsrc[i];
    }
    __syncthreads();

    const int colTile = blockIdx.x * WAVES_PER_BLOCK + wave;
    if (colTile < colTiles && (colTile * 16 + 16) <= V) {   // wave-uniform -> EXEC all 1s
        const int col0 = colTile * 16;
        v8f c = {};
        const float*  brow   = table + (size_t)(col0 + l16) * D;  // B[K][N]: N=l16 row of table
        const float2* a2base = (const float2*)(Atile + l16 * D);  // A[M][K]: M=l16 row of gen
        #pragma unroll 4
        for (int k0 = 0; k0 < D; k0 += 4) {
            // lane holds K = k0 + 2*half, k0 + 2*half + 1 (A layout; B mirrored)
            float2 af = a2base[(k0 >> 1) + half];
            float2 bf = *(const float2*)(brow + k0 + 2 * half);
            v2f a; a.x = af.x; a.y = af.y;
            v2f b; b.x = bf.x; b.y = bf.y;
            c = __builtin_amdgcn_wmma_f32_16x16x4_f32(
                    false, a, false, b, (short)0, c, false, false);
        }
        // C/D layout: VGPR r -> M = r + 8*half, N = l16
        size_t outBase = (size_t)(row0 + 8 * half) * V + col0 + l16;
        #pragma unroll
        for (int r = 0; r < 8; ++r) {
            scores[outBase + (size_t)r * V] = c[r];
        }
    }
}

// ---------------------------------------------------------------------------
// Kernel 2: scatter NEG_INF over each row's seed-track columns.
// ---------------------------------------------------------------------------
__global__ void mask_seeds(const int* __restrict__ seeds,
                           float* __restrict__ scores,
                           int rows, int S, int V) {
    int i = blockIdx.x * blockDim.x + threadIdx.x;
    if (i < rows * S) {
        int b = i / S;
        int v = seeds[i];
        if (v >= 0 && v < V) scores[(size_t)b * V + v] = -1e30f;
    }
}

// ---------------------------------------------------------------------------
// Kernel 3: per-row top-500, sorted descending (ties: lower index first).
// 4-pass 8-bit radix select on monotone uint keys, then bitonic sort of 512.
// One 256-thread block per row; row data served from L2.
// ---------------------------------------------------------------------------
__device__ __forceinline__ unsigned flip_f32(float f) {
    unsigned u = __float_as_uint(f);
    return ((int)u < 0) ? ~u : (u | 0x80000000u);   // monotone: larger f -> larger u
}

__global__ void topk_row(const float* __restrict__ scores,
                         float* __restrict__ outVals,
                         float* __restrict__ outIdx,
                         int V, int K) {
    const int b   = blockIdx.x;
    const int tid = threadIdx.x;
    const float* row = scores + (size_t)b * V;

    __shared__ unsigned hist[256];
    __shared__ unsigned sPrefix, sKrem, sCntG, sCntE;
    __shared__ unsigned long long keys[SORT_N];

    if (tid == 0) { sPrefix = 0u; sKrem = (unsigned)K; }

    // --- radix select: find exact value T of the K-th largest element ---
    for (int round = 0; round < 4; ++round) {
        const int shift = 24 - 8 * round;
        const unsigned maskHi = (round == 0) ? 0u : (0xFFFFFFFFu << (shift + 8));
        for (int i = tid; i < 256; i += blockDim.x) hist[i] = 0u;
        __syncthreads();
        const unsigned prefix = sPrefix;
        for (int v = tid; v < V; v += blockDim.x) {
            unsigned u = flip_f32(row[v]);
            if ((u & maskHi) == (prefix & maskHi))
                atomicAdd(&hist[(u >> shift) & 0xFF], 1u);
        }
        __syncthreads();
        if (tid == 0) {
            unsigned k = sKrem, cum = 0;
            int bsel = 0;
            for (int bb = 255; bb >= 0; --bb) {
                if (cum + hist[bb] >= k) { bsel = bb; break; }
                cum += hist[bb];
            }
            sKrem   = k - cum;                       // # of ties (== T) to keep
            sPrefix = prefix | ((unsigned)bsel << shift);
        }
        __syncthreads();
    }
    const unsigned T          = sPrefix;
    const unsigned numEq      = sKrem;
    const unsigned numGreater = (unsigned)K - numEq;

    if (tid == 0) { sCntG = 0u; sCntE = 0u; }
    for (int i = tid; i < SORT_N; i += blockDim.x) keys[i] = 0ull;
    __syncthreads();

    // --- collect the K winners: composite key (flipped_val << 32) | ~idx ---
    for (int v = tid; v < V; v += blockDim.x) {
        unsigned u = flip_f32(row[v]);
        if (u > T) {
            unsigned pos = atomicAdd(&sCntG, 1u);
            if (pos < (unsigned)K)
                keys[pos] = ((unsigned long long)u << 32) | (unsigned)~(unsigned)v;
        } else if (u == T) {
            unsigned e = atomicAdd(&sCntE, 1u);
            if (e < numEq) {
                unsigned pos = numGreater + e;
                if (pos < (unsigned)K)
                    keys[pos] = ((unsigned long long)u << 32) | (unsigned)~(unsigned)v;
            }
        }
    }
    __syncthreads();

    // --- bitonic sort 512 keys, descending ---
    for (unsigned ksz = 2; ksz <= SORT_N; ksz <<= 1) {
        for (unsigned j = ksz >> 1; j > 0; j >>= 1) {
            for (unsigned t = tid; t < SORT_N; t += blockDim.x) {
                unsigned partner = t ^ j;
                if (partner > t) {
                    unsigned long long a = keys[t];
                    unsigned long long c = keys[partner];
                    bool desc = ((t & ksz) == 0);
                    if (desc ? (a < c) : (a > c)) { keys[t] = c; keys[partner] = a; }
                }
            }
            __syncthreads();
        }
    }

    // --- emit values and indices (indices as floats, exact for V < 2^24) ---
    for (int i = tid; i < K; i += blockDim.x) {
        unsigned long long kk = keys[i];
        unsigned u   = (unsigned)(kk >> 32);
        unsigned fb  = (u & 0x80000000u) ? (u ^ 0x80000000u) : ~u;  // un-flip
        unsigned idx = ~(unsigned)(kk & 0xFFFFFFFFu);
        outVals[(size_t)b * K + i] = __uint_as_float(fb);
        outIdx [(size_t)b * K + i] = (float)idx;
    }
}

// ---------------------------------------------------------------------------
extern "C" void kernel_launch(void* const* d_in, const int* in_sizes, int n_in,
                              void* d_out, int out_size, void* d_ws, size_t ws_size,
                              hipStream_t stream) {
    const float* gen   = (const float*)d_in[0];
    const int*   seeds = (const int*)d_in[1];
    const float* table = (const float*)d_in[2];

    const int D = 128;
    const int B = in_sizes[0] / D;          // 1024
    const int V = in_sizes[2] / D;          // 100000
    const int S = in_sizes[1] / B;          // 100
    const int K = TOPK;

    float* outVals = (float*)d_out;
    float* outIdx  = outVals + (size_t)B * K;
    float* scores  = (float*)d_ws;

    const size_t rowBytes = (size_t)V * sizeof(float);
    int chunkRows = (ws_size >= rowBytes) ? (int)(ws_size / rowBytes) : 16;
    chunkRows &= ~15;                        // multiple of 16 (row-tile height)
    if (chunkRows < 16) chunkRows = 16;
    if (chunkRows > B)  chunkRows = B;

    const int colTiles = (V + 15) / 16;
    const int gridX    = (colTiles + WAVES_PER_BLOCK - 1) / WAVES_PER_BLOCK;

    for (int rowBase = 0; rowBase < B; rowBase += chunkRows) {
        int rows = B - rowBase;
        if (rows > chunkRows) rows = chunkRows;

        dim3 ggrid(gridX, rows / 16);
        gemm_wmma_f32<<<ggrid, dim3(256), 0, stream>>>(
            gen + (size_t)rowBase * D, table, scores, V, D, colTiles);

        int mthreads = rows * S;
        mask_seeds<<<(mthreads + 255) / 256, 256, 0, stream>>>(
            seeds + (size_t)rowBase * S, scores, rows, S, V);

        topk_row<<<rows, 256, 0, stream>>>(
            scores, outVals + (size_t)rowBase * K, outIdx + (size_t)rowBase * K, V, K);
    }
}